// JacobiMachine_81836306858790
// MI455X (gfx1250) — compile-verified
//
#include <hip/hip_runtime.h>
#include <cmath>

typedef float v2f __attribute__((ext_vector_type(2)));
typedef float v8f __attribute__((ext_vector_type(8)));

#define NXDIM 4096
#define NTILE 256   // NXDIM / 16
#define NSTEP 100

// ---------------------------------------------------------------------------
// Init: x0 = exp(-50*((X-0.5)^2 + (Y-0.5)^2)), 4 elements per thread.
// ---------------------------------------------------------------------------
__global__ __launch_bounds__(256) void jacobi_init(const float* __restrict__ X,
                                                   const float* __restrict__ Y,
                                                   float* __restrict__ out) {
    size_t i = (size_t)blockIdx.x * 256 + threadIdx.x;
    float4 x = ((const float4*)X)[i];
    float4 y = ((const float4*)Y)[i];
    float4 r;
    r.x = expf(-50.0f * ((x.x - 0.5f) * (x.x - 0.5f) + (y.x - 0.5f) * (y.x - 0.5f)));
    r.y = expf(-50.0f * ((x.y - 0.5f) * (x.y - 0.5f) + (y.y - 0.5f) * (y.y - 0.5f)));
    r.z = expf(-50.0f * ((x.z - 0.5f) * (x.z - 0.5f) + (y.z - 0.5f) * (y.z - 0.5f)));
    r.w = expf(-50.0f * ((x.w - 0.5f) * (x.w - 0.5f) + (y.w - 0.5f) * (y.w - 0.5f)));
    ((float4*)out)[i] = r;
}

// ---------------------------------------------------------------------------
// One Jacobi sweep via WMMA:  X' = mask .* ( T*X + X*T ),  T = 0.25*tridiag.
// One wave32 computes one 16x16 output tile with 12 v_wmma_f32_16x16x4_f32.
// ---------------------------------------------------------------------------
__device__ __forceinline__ v8f wmma_acc(v2f a, v2f b, v8f c) {
    // 8 args: (neg_a, A, neg_b, B, c_mod, C, reuse_a, reuse_b)
    return __builtin_amdgcn_wmma_f32_16x16x4_f32(false, a, false, b, (short)0, c,
                                                 false, false);
}

__global__ __launch_bounds__(128) void jacobi_step(const float* __restrict__ in,
                                                   float* __restrict__ out) {
    const int lane = threadIdx.x & 31;
    const int wave = threadIdx.x >> 5;
    const int tJ = blockIdx.x * 4 + wave;     // tile column, wave-uniform
    const int tI = blockIdx.y;                // tile row,    wave-uniform
    const int m  = lane & 15;
    const bool hi = lane >= 16;

    const int row0 = tI * 16;
    const int col0 = tJ * 16;

    v8f acc = {0.f, 0.f, 0.f, 0.f, 0.f, 0.f, 0.f, 0.f};

    // ================= Row direction:  acc += G_t * X[tI+t, tJ] =================
    // B-fragment of an X tile, chunk c (kb = 4c + (hi?2:0)):
    //   comp0 = X[kb, n], comp1 = X[kb+1, n], n = lane%16.
    // Center tile: G_mid(m,k) = 0.25 if |m-k|==1 — all 4 chunks.
    #pragma unroll
    for (int c = 0; c < 4; ++c) {
        const int kb = 4 * c + (hi ? 2 : 0);
        const float* p = in + (size_t)(row0 + kb) * NXDIM + (col0 + m);
        v2f b; b.x = p[0]; b.y = p[NXDIM];
        v2f a;
        a.x = (kb     == m - 1 || kb     == m + 1) ? 0.25f : 0.0f;
        a.y = (kb + 1 == m - 1 || kb + 1 == m + 1) ? 0.25f : 0.0f;
        acc = wmma_acc(a, b, acc);
    }
    if (tI > 0) {
        // Up coupling G_up[0,15]=0.25 lives in chunk c=3 only (K=12..15).
        const int kb = 12 + (hi ? 2 : 0);
        const float* p = in + (size_t)(row0 - 16 + kb) * NXDIM + (col0 + m);
        v2f b; b.x = p[0]; b.y = p[NXDIM];
        v2f a;
        a.x = 0.0f;                                   // K=12 or 14: never 15
        a.y = (hi && m == 0) ? 0.25f : 0.0f;          // K=15 row 0
        acc = wmma_acc(a, b, acc);
    }
    if (tI < NTILE - 1) {
        // Down coupling G_dn[15,0]=0.25 lives in chunk c=0 only (K=0..3).
        const int kb = (hi ? 2 : 0);
        const float* p = in + (size_t)(row0 + 16 + kb) * NXDIM + (col0 + m);
        v2f b; b.x = p[0]; b.y = p[NXDIM];
        v2f a;
        a.x = (!hi && m == 15) ? 0.25f : 0.0f;        // K=0 row 15
        a.y = 0.0f;
        acc = wmma_acc(a, b, acc);
    }

    // ================= Column direction:  acc += X[tI, tJ+t] * R_t =============
    // A-fragment of an X tile, chunk c: lane m holds {X[m,kb], X[m,kb+1]} (b64).
    // Center tile: R_mid(k,n) = 0.25 if |k-n|==1 — all 4 chunks.
    #pragma unroll
    for (int c = 0; c < 4; ++c) {
        const int kb = 4 * c + (hi ? 2 : 0);
        const float2 v = *(const float2*)(in + (size_t)(row0 + m) * NXDIM + (col0 + kb));
        v2f a; a.x = v.x; a.y = v.y;
        v2f b;
        b.x = (kb     == m - 1 || kb     == m + 1) ? 0.25f : 0.0f;  // R[kb,   n]
        b.y = (kb + 1 == m - 1 || kb + 1 == m + 1) ? 0.25f : 0.0f;  // R[kb+1, n]
        acc = wmma_acc(a, b, acc);
    }
    if (tJ > 0) {
        // Left coupling R_l[15,0]=0.25 → chunk c=3 only.
        const int kb = 12 + (hi ? 2 : 0);
        const float2 v = *(const float2*)(in + (size_t)(row0 + m) * NXDIM + (col0 - 16 + kb));
        v2f a; a.x = v.x; a.y = v.y;
        v2f b;
        b.x = 0.0f;
        b.y = (hi && m == 0) ? 0.25f : 0.0f;          // R[15, n]: n==0
        acc = wmma_acc(a, b, acc);
    }
    if (tJ < NTILE - 1) {
        // Right coupling R_r[0,15]=0.25 → chunk c=0 only.
        const int kb = (hi ? 2 : 0);
        const float2 v = *(const float2*)(in + (size_t)(row0 + m) * NXDIM + (col0 + 16 + kb));
        v2f a; a.x = v.x; a.y = v.y;
        v2f b;
        b.x = (!hi && m == 15) ? 0.25f : 0.0f;        // R[0, n]: n==15
        b.y = 0.0f;
        acc = wmma_acc(a, b, acc);
    }

    // ============ Store with zero-Dirichlet boundary mask ============
    // C/D layout: VGPR r -> row = r + (hi?8:0), col = lane%16.
    const int colg = col0 + m;
    const bool colEdge = (colg == 0) || (colg == NXDIM - 1);
    #pragma unroll
    for (int r = 0; r < 8; ++r) {
        const int rowg = row0 + r + (hi ? 8 : 0);
        float v = acc[r];
        if (colEdge || rowg == 0 || rowg == NXDIM - 1) v = 0.0f;
        out[(size_t)rowg * NXDIM + colg] = v;
    }
}

// ---------------------------------------------------------------------------
// Host side: init into d_out, then 100 ping-pong sweeps d_out <-> d_ws.
// 100 is even, so the final sweep (t=99: ws -> out) lands in d_out.
// ---------------------------------------------------------------------------
extern "C" void kernel_launch(void* const* d_in, const int* in_sizes, int n_in,
                              void* d_out, int out_size, void* d_ws, size_t ws_size,
                              hipStream_t stream) {
    const float* X = (const float*)d_in[0];
    const float* Y = (const float*)d_in[1];
    float* bufA = (float*)d_out;
    float* bufB = (float*)d_ws;   // needs NXDIM*NXDIM*4 = 64 MB of scratch

    const size_t total = (size_t)NXDIM * NXDIM;
    jacobi_init<<<dim3((unsigned)(total / 4 / 256)), dim3(256), 0, stream>>>(X, Y, bufA);

    dim3 grid(NTILE / 4, NTILE);
    dim3 block(128);
    for (int t = 0; t < NSTEP; ++t) {
        const float* src = (t & 1) ? bufB : bufA;
        float*       dst = (t & 1) ? bufA : bufB;
        jacobi_step<<<grid, block, 0, stream>>>(src, dst);
    }
}